// QuantumValueEncoder_8272107012583
// MI455X (gfx1250) — compile-verified
//
#include <hip/hip_runtime.h>
#include <math.h>

#define NQ 19
#define NSTATE (1 << NQ)
#define NL 3
#define NA 8
#define PI_F 3.14159265358979323846f
#define TWOPI_F 6.28318530717958647692f

typedef float v2f __attribute__((ext_vector_type(2)));
typedef float v8f __attribute__((ext_vector_type(8)));

__device__ __forceinline__ float2 cmulf(float2 a, float2 b) {
  return make_float2(a.x * b.x - a.y * b.y, a.x * b.y + a.y * b.x);
}

// Insert zero bits at sorted positions q0<q1<q2<q3 (scatter 15 outer bits into 19-bit index).
__device__ __forceinline__ int expand_outer(int o, int q0, int q1, int q2, int q3) {
  int x = o;
  x = ((x >> q0) << (q0 + 1)) | (x & ((1 << q0) - 1));
  x = ((x >> q1) << (q1 + 1)) | (x & ((1 << q1) - 1));
  x = ((x >> q2) << (q2 + 1)) | (x & ((1 << q2) - 1));
  x = ((x >> q3) << (q3 + 1)) | (x & ((1 << q3) - 1));
  return x;
}

// Deposit 4-bit group combo j into bit positions q0..q3.
__device__ __forceinline__ int jdep(int j, int q0, int q1, int q2, int q3) {
  return ((j & 1) << q0) | (((j >> 1) & 1) << q1) | (((j >> 2) & 1) << q2) | (((j >> 3) & 1) << q3);
}

__global__ void init_state(float2* st) {
  int i = blockIdx.x * blockDim.x + threadIdx.x;
  if (i < NSTATE) st[i] = make_float2(i == 0 ? 1.0f : 0.0f, 0.0f);
}

// Build per-(layer,wire) fused 2x2 U = RZ * RY * RX, then 16x16 kron matrices per (layer,group).
// mats layout: [(l*5+g)] -> 512 floats: [0..255] = Re(U16[i*16+j]), [256..511] = Im.
__global__ void prep_gates(const float* __restrict__ xin, const float* __restrict__ yw,
                           const float* __restrict__ zw, float* __restrict__ mats) {
  __shared__ float2 G[NL][NQ][2][2];
  int t = threadIdx.x;
  if (t < NL * NQ) {
    int w = t % NQ;
    float x = xin[w];
    float ang;
    if (w >= NQ - 2) {                       // directional features -> {0, pi}
      ang = (x >= 0.5f) ? PI_F : 0.0f;
    } else {                                 // (x - LO)/span * 2pi, LO=-1.5 span=3
      ang = (x + 1.5f) * (TWOPI_F / 3.0f);
    }
    ang = fminf(fmaxf(ang, 0.0f), TWOPI_F);
    float cx = cosf(0.5f * ang), sx = sinf(0.5f * ang);
    // RX = [[(cx,0),(0,-sx)],[(0,-sx),(cx,0)]]
    float2 rx00 = make_float2(cx, 0.0f);
    float2 rx01 = make_float2(0.0f, -sx);    // rx10 == rx01, rx11 == rx00
    float cy = cosf(0.5f * yw[t]), sy = sinf(0.5f * yw[t]);
    // M = RY * RX, RY = [[cy,-sy],[sy,cy]] (real)
    float2 m00 = make_float2(cy * rx00.x - sy * rx01.x, cy * rx00.y - sy * rx01.y);
    float2 m01 = make_float2(cy * rx01.x - sy * rx00.x, cy * rx01.y - sy * rx00.y);
    float2 m10 = make_float2(sy * rx00.x + cy * rx01.x, sy * rx00.y + cy * rx01.y);
    float2 m11 = make_float2(sy * rx01.x + cy * rx00.x, sy * rx01.y + cy * rx00.y);
    float cz = cosf(0.5f * zw[t]), sz = sinf(0.5f * zw[t]);
    float2 d0 = make_float2(cz, -sz);        // exp(-i z/2)
    float2 d1 = make_float2(cz, sz);         // exp(+i z/2)
    int l = t / NQ;
    G[l][w][0][0] = cmulf(d0, m00);
    G[l][w][0][1] = cmulf(d0, m01);
    G[l][w][1][0] = cmulf(d1, m10);
    G[l][w][1][1] = cmulf(d1, m11);
  }
  __syncthreads();

  const int GRP[5][4] = {{0,1,2,3},{4,5,6,7},{8,9,10,11},{12,13,14,15},{15,16,17,18}};
  int i = t >> 4, j = t & 15;
  for (int l = 0; l < NL; ++l) {
    for (int g = 0; g < 5; ++g) {
      float2 acc = make_float2(1.0f, 0.0f);
      #pragma unroll
      for (int k = 0; k < 4; ++k) {
        int bit = GRP[g][k];
        int ik = (i >> k) & 1, jk = (j >> k) & 1;
        float2 u;
        if (g == 4 && k == 0) {              // pad slot (bit 15): identity this pass
          u = make_float2((ik == jk) ? 1.0f : 0.0f, 0.0f);
        } else {
          int wire = 18 - bit;               // reference wire w <-> flat-index bit 18-w
          u = G[l][wire][ik][jk];
        }
        acc = cmulf(acc, u);
      }
      float* m = mats + (l * 5 + g) * 512;
      m[i * 16 + j] = acc.x;
      m[256 + i * 16 + j] = acc.y;
    }
  }
}

// Apply one 16x16 complex unitary to 4 gathered qubits, in place, via V_WMMA_F32_16X16X4_F32.
// Each wave: 16 outer indices x 16 combos = 256 amplitudes; complex GEMM = 16 wmma ops.
__global__ void gate_pass(float2* __restrict__ st, const float* __restrict__ mats,
                          int q0, int q1, int q2, int q3, int applyCZ) {
  const int lane = threadIdx.x & 31;
  const int wave = (blockIdx.x * blockDim.x + threadIdx.x) >> 5;
  const int col = lane & 15;                 // outer slot within the wave tile
  const int half = lane >> 4;                // lanes 16-31 hold K/row +2 halves (ISA 7.12.2)
  const int o = wave * 16 + col;
  const int base = expand_outer(o, q0, q1, q2, q3);

  const float* Ure = mats;
  const float* Uim = mats + 256;

  v2f ar[4], ai[4], br[4], bi[4];
  #pragma unroll
  for (int k = 0; k < 4; ++k) {
    int K = 4 * k + 2 * half;                // A 16x4 layout: lanes<16 K={0,1}, lanes>=16 K={2,3}
    ar[k].x = Ure[col * 16 + K];  ar[k].y = Ure[col * 16 + K + 1];
    ai[k].x = Uim[col * 16 + K];  ai[k].y = Uim[col * 16 + K + 1];
    // B 4x16: same half split for rows; column = outer slot
    float2 s0 = st[base + jdep(K,     q0, q1, q2, q3)];
    float2 s1 = st[base + jdep(K + 1, q0, q1, q2, q3)];
    br[k].x = s0.x;  br[k].y = s1.x;
    bi[k].x = s0.y;  bi[k].y = s1.y;
  }

  v8f dr = {0.f,0.f,0.f,0.f,0.f,0.f,0.f,0.f};
  v8f di = {0.f,0.f,0.f,0.f,0.f,0.f,0.f,0.f};
  #pragma unroll
  for (int k = 0; k < 4; ++k) {
    v2f ani;                                 // f32 WMMA NEG is CNeg-only -> negate U_i in VALU
    ani.x = -ai[k].x;  ani.y = -ai[k].y;
    dr = __builtin_amdgcn_wmma_f32_16x16x4_f32(false, ar[k], false, br[k], (short)0, dr, false, false);
    dr = __builtin_amdgcn_wmma_f32_16x16x4_f32(false, ani,   false, bi[k], (short)0, dr, false, false);
    di = __builtin_amdgcn_wmma_f32_16x16x4_f32(false, ar[k], false, bi[k], (short)0, di, false, false);
    di = __builtin_amdgcn_wmma_f32_16x16x4_f32(false, ai[k], false, br[k], (short)0, di, false, false);
  }

  #pragma unroll
  for (int r = 0; r < 8; ++r) {
    int row = r + 8 * half;                  // D 16x16: reg r -> rows r / r+8
    int idx = base + jdep(row, q0, q1, q2, q3);
    float re = dr[r], im = di[r];
    if (applyCZ) {                           // CZ ring diagonal: pairs (b,b+1) b=0..17 and (0,18)
      int p = __popc(idx & (idx >> 1)) + ((idx >> 18) & idx & 1);
      if (p & 1) { re = -re; im = -im; }
    }
    st[idx] = make_float2(re, im);
  }
}

// Stage 1: per-block partial <Z_w> sums (wire w -> flat bit 18-w), w=0..7.
__global__ void reduce1(const float2* __restrict__ st, float* __restrict__ partials) {
  const int tid = threadIdx.x;
  const int gid = blockIdx.x * 256 + tid;
  float acc[NA];
  #pragma unroll
  for (int w = 0; w < NA; ++w) acc[w] = 0.0f;
  #pragma unroll
  for (int it = 0; it < NSTATE / 65536; ++it) {
    int idx = gid + it * 65536;
    float2 a = st[idx];
    float p = a.x * a.x + a.y * a.y;
    #pragma unroll
    for (int w = 0; w < NA; ++w) {
      acc[w] += ((idx >> (18 - w)) & 1) ? -p : p;
    }
  }
  __shared__ float sd[256];
  for (int w = 0; w < NA; ++w) {
    sd[tid] = acc[w];
    __syncthreads();
    for (int s = 128; s > 0; s >>= 1) {
      if (tid < s) sd[tid] += sd[tid + s];
      __syncthreads();
    }
    if (tid == 0) partials[blockIdx.x * NA + w] = sd[0];
    __syncthreads();
  }
}

// Stage 2: fold 256 partials per output (deterministic tree).
__global__ void reduce2(const float* __restrict__ partials, float* __restrict__ out) {
  __shared__ float sd[256];
  const int tid = threadIdx.x;
  for (int w = 0; w < NA; ++w) {
    sd[tid] = partials[tid * NA + w];
    __syncthreads();
    for (int s = 128; s > 0; s >>= 1) {
      if (tid < s) sd[tid] += sd[tid + s];
      __syncthreads();
    }
    if (tid == 0) out[w] = sd[0];
    __syncthreads();
  }
}

extern "C" void kernel_launch(void* const* d_in, const int* in_sizes, int n_in,
                              void* d_out, int out_size, void* d_ws, size_t ws_size,
                              hipStream_t stream) {
  (void)in_sizes; (void)n_in; (void)out_size; (void)ws_size;
  const float* xin = (const float*)d_in[0];   // inputs [19]
  const float* yw  = (const float*)d_in[1];   // y_weights [3,19]
  const float* zw  = (const float*)d_in[2];   // z_weights [3,19]
  float* out = (float*)d_out;                 // [8] f32

  float* wsf      = (float*)d_ws;
  float* mats     = wsf;                      // 15 * 512 floats
  float* partials = wsf + 8192;               // 256 * 8 floats
  float2* st      = (float2*)(wsf + 16384);   // 2^19 complex64 (4 MiB)

  init_state<<<NSTATE / 256, 256, 0, stream>>>(st);
  prep_gates<<<1, 256, 0, stream>>>(xin, yw, zw, mats);

  const int GRP[5][4] = {{0,1,2,3},{4,5,6,7},{8,9,10,11},{12,13,14,15},{15,16,17,18}};
  for (int l = 0; l < NL; ++l) {
    for (int g = 0; g < 5; ++g) {
      gate_pass<<<256, 256, 0, stream>>>(st, mats + (l * 5 + g) * 512,
                                         GRP[g][0], GRP[g][1], GRP[g][2], GRP[g][3],
                                         (g == 4) ? 1 : 0);
    }
  }

  reduce1<<<256, 256, 0, stream>>>(st, partials);
  reduce2<<<1, 256, 0, stream>>>(partials, out);
}